// VisionTraceAggregator_13520557048022
// MI455X (gfx1250) — compile-verified
//
#include <hip/hip_runtime.h>
#include <hip/hip_bf16.h>

// Problem constants (match reference)
constexpr int B_  = 32;
constexpr int T_  = 2000;
constexpr int H_  = 768;
constexpr int S_  = 100;
constexpr int NV_ = 100;
constexpr int K2_ = 2 * H_;   // 1536

typedef float v2f __attribute__((ext_vector_type(2)));
typedef float v8f __attribute__((ext_vector_type(8)));

// ---------------------------------------------------------------------------
// Kernel 1: segment mean.  One block per (b, s) segment; scans the mask row
// (uniform -> scalar loads) and accumulates matching trace rows in registers.
// Each trace row belongs to exactly one segment, so total HBM traffic for the
// big tensor is exactly one pass (~196 MB) with zero atomics.
// ---------------------------------------------------------------------------
__global__ __launch_bounds__(256)
void seg_mean_kernel(const float* __restrict__ feat,   // (B, NV+T, H)
                     const int*   __restrict__ mask,   // (B, T)
                     float*       __restrict__ aggs)   // (B*S, H)
{
    const int seg = blockIdx.x;          // 0 .. B*S-1
    const int b   = seg / S_;
    const int sv  = (seg % S_) + 1;      // mask value for this segment
    const int tid = threadIdx.x;         // 0..255, H/256 = 3 elems per thread

    const float* base = feat + ((size_t)b * (NV_ + T_) + NV_) * H_;
    const int*   mrow = mask + (size_t)b * T_;

    float a0 = 0.f, a1 = 0.f, a2 = 0.f;
    int count = 0;
    for (int t = 0; t < T_; ++t) {
        if (mrow[t] == sv) {
            const float* row = base + (size_t)t * H_;
            a0 += row[tid];
            a1 += row[tid + 256];
            a2 += row[tid + 512];
            ++count;
        }
    }
    const float inv = 1.0f / (float)count;   // count==0 -> inf, same as reference
    float* o = aggs + (size_t)seg * H_;
    o[tid]       = a0 * inv;
    o[tid + 256] = a1 * inv;
    o[tid + 512] = a2 * inv;
}

// ---------------------------------------------------------------------------
// Kernel 2: vision mean over first NV tokens.  One block per batch.
// ---------------------------------------------------------------------------
__global__ __launch_bounds__(256)
void vision_mean_kernel(const float* __restrict__ feat,   // (B, NV+T, H)
                        float*       __restrict__ vfeat)  // (B, H)
{
    const int b   = blockIdx.x;
    const int tid = threadIdx.x;
    const float* base = feat + (size_t)b * (NV_ + T_) * H_;

    float a0 = 0.f, a1 = 0.f, a2 = 0.f;
    for (int t = 0; t < NV_; ++t) {
        const float* row = base + (size_t)t * H_;
        a0 += row[tid];
        a1 += row[tid + 256];
        a2 += row[tid + 512];
    }
    const float inv = 1.0f / (float)NV_;
    float* o = vfeat + (size_t)b * H_;
    o[tid]       = a0 * inv;
    o[tid + 256] = a1 * inv;
    o[tid + 512] = a2 * inv;
}

// ---------------------------------------------------------------------------
// Kernel 3: merge GEMM with f32 WMMA.
// out(M=3200, N=768) = vt(M, K=1536) * w_merge^T + bias,
//   vt[:, :768]  = aggs,  vt[:, 768:] = vfeat[row/S]  (never materialized).
// One wave per 16x64 output block; 4 accumulators; K stepped by 4 with
// V_WMMA_F32_16X16X4_F32.  Fragment layouts per CDNA5 ISA 7.12.2:
//   A (16x4 f32):  lane l, vgpr v -> A[l&15][2*(l>>4)+v]
//   B (4x16 f32):  lane l, vgpr v -> B[2*(l>>4)+v][l&15]
//   C/D (16x16):   lane l, vgpr v -> D[v + 8*(l>>4)][l&15]
// ---------------------------------------------------------------------------
#if __has_builtin(__builtin_amdgcn_wmma_f32_16x16x4_f32)
#define HAVE_WMMA_F32X4 1
#else
#define HAVE_WMMA_F32X4 0
#endif

#if HAVE_WMMA_F32X4
__global__ __launch_bounds__(32)
void merge_gemm_wmma_kernel(const float* __restrict__ aggs,   // (3200, 768)
                            const float* __restrict__ vfeat,  // (32, 768)
                            const float* __restrict__ w,      // (768, 1536) row-major
                            const float* __restrict__ bias,   // (768)
                            float*       __restrict__ out)    // (3200, 768)
{
    const int lane  = threadIdx.x;        // 0..31, one wave per block
    const int m0    = blockIdx.x * 16;    // 200 m-tiles
    const int n0    = blockIdx.y * 64;    // 12 n-supertiles
    const int rrow  = lane & 15;
    const int halfk = (lane >> 4) << 1;   // 0 or 2

    const int m = m0 + rrow;                           // A-fragment row
    const float* arow_agg = aggs  + (size_t)m * H_;
    const float* arow_vis = vfeat + (size_t)(m / S_) * H_;

    const float* w0 = w + (size_t)(n0 + rrow)      * K2_;
    const float* w1 = w + (size_t)(n0 + 16 + rrow) * K2_;
    const float* w2 = w + (size_t)(n0 + 32 + rrow) * K2_;
    const float* w3 = w + (size_t)(n0 + 48 + rrow) * K2_;

    v8f acc0 = {}; v8f acc1 = {}; v8f acc2 = {}; v8f acc3 = {};

    for (int k = 0; k < K2_; k += 4) {
        const int koff = k + halfk;                    // even -> 8B aligned
        v2f a = (k < H_) ? *(const v2f*)(arow_agg + koff)
                         : *(const v2f*)(arow_vis + (koff - H_));
        v2f b0 = *(const v2f*)(w0 + koff);
        v2f b1 = *(const v2f*)(w1 + koff);
        v2f b2 = *(const v2f*)(w2 + koff);
        v2f b3 = *(const v2f*)(w3 + koff);

        acc0 = __builtin_amdgcn_wmma_f32_16x16x4_f32(false, a, false, b0, (short)0, acc0, false, false);
        acc1 = __builtin_amdgcn_wmma_f32_16x16x4_f32(false, a, false, b1, (short)0, acc1, false, false);
        acc2 = __builtin_amdgcn_wmma_f32_16x16x4_f32(false, a, false, b2, (short)0, acc2, false, false);
        acc3 = __builtin_amdgcn_wmma_f32_16x16x4_f32(false, a, false, b3, (short)0, acc3, false, false);
    }

    // Write back with bias.  D row = v + 8*(lane>>4), col = rrow.
    const int orow0 = m0 + ((lane >> 4) << 3);
    const float bv0 = bias[n0 + rrow];
    const float bv1 = bias[n0 + 16 + rrow];
    const float bv2 = bias[n0 + 32 + rrow];
    const float bv3 = bias[n0 + 48 + rrow];
    #pragma unroll
    for (int v = 0; v < 8; ++v) {
        float* orow = out + (size_t)(orow0 + v) * H_;
        orow[n0 +      rrow] = acc0[v] + bv0;
        orow[n0 + 16 + rrow] = acc1[v] + bv1;
        orow[n0 + 32 + rrow] = acc2[v] + bv2;
        orow[n0 + 48 + rrow] = acc3[v] + bv3;
    }
}
#else
// Fallback (only if the f32 WMMA builtin is missing): naive per-element dot.
__global__ __launch_bounds__(256)
void merge_gemm_naive_kernel(const float* __restrict__ aggs,
                             const float* __restrict__ vfeat,
                             const float* __restrict__ w,
                             const float* __restrict__ bias,
                             float*       __restrict__ out)
{
    const int n = blockIdx.x * blockDim.x + threadIdx.x;   // 0..767
    const int m = blockIdx.y;                              // 0..3199
    if (n >= H_) return;
    const float* wrow = w + (size_t)n * K2_;
    const float* arow = aggs + (size_t)m * H_;
    const float* vrow = vfeat + (size_t)(m / S_) * H_;
    float acc = bias[n];
    for (int k = 0; k < H_; ++k) acc += arow[k] * wrow[k];
    for (int k = 0; k < H_; ++k) acc += vrow[k] * wrow[H_ + k];
    out[(size_t)m * H_ + n] = acc;
}
#endif

// ---------------------------------------------------------------------------
extern "C" void kernel_launch(void* const* d_in, const int* in_sizes, int n_in,
                              void* d_out, int out_size, void* d_ws, size_t ws_size,
                              hipStream_t stream) {
    const float* feat = (const float*)d_in[0];   // (B, NV+T, H) f32
    const int*   mask = (const int*)  d_in[1];   // (B, T) i32
    const float* w    = (const float*)d_in[2];   // (H, 2H) f32
    const float* bias = (const float*)d_in[3];   // (H,) f32
    float*       out  = (float*)d_out;           // (B, S, H) f32

    float* aggs  = (float*)d_ws;                         // B*S*H floats
    float* vfeat = aggs + (size_t)B_ * S_ * H_;          // B*H floats

    seg_mean_kernel<<<dim3(B_ * S_), 256, 0, stream>>>(feat, mask, aggs);
    vision_mean_kernel<<<dim3(B_), 256, 0, stream>>>(feat, vfeat);

#if HAVE_WMMA_F32X4
    merge_gemm_wmma_kernel<<<dim3((B_ * S_) / 16, H_ / 64), 32, 0, stream>>>(
        aggs, vfeat, w, bias, out);
#else
    merge_gemm_naive_kernel<<<dim3(H_ / 256, B_ * S_), 256, 0, stream>>>(
        aggs, vfeat, w, bias, out);
#endif
}